// MemoryModule_3994319585527
// MI455X (gfx1250) — compile-verified
//
#include <hip/hip_runtime.h>

typedef __attribute__((ext_vector_type(2))) float v2f;
typedef __attribute__((ext_vector_type(8))) float v8f;

#define B_   2
#define T_   4
#define CK_  64
#define CV_  256
#define H_   32
#define W_   64
#define N_   2048      // H*W  (query locations)
#define M_   8192      // T*H*W (memory locations)
#define NT_  32        // query tile per workgroup
#define MC_  64        // memory chunk per iteration
#define NCHUNK_ (M_/MC_)

#define QT_S 68        // Qt  [n'][ck]   stride (16B-aligned rows)
#define K_S  68        // K   [ck][w]    stride (16B-aligned rows)
#define V_S  68        // V   [cv][w]    stride (16B-aligned rows)
#define ST_S 66        // S^T [n'][m]    stride (8B-aligned rows)

#define ASYNC_PER_CHUNK 20   // per-thread async b128 issues per chunk (4 K + 16 V)

#define LDS_OFF(p) ((unsigned int)(uintptr_t)(p))

__device__ __forceinline__ v8f wmma4(v2f a, v2f b, v8f c) {
  // D = A(16x4,f32) * B(4x16,f32) + C(16x16,f32)
  return __builtin_amdgcn_wmma_f32_16x16x4_f32(false, a, false, b, (short)0, c,
                                               false, false);
}

// 16B global -> LDS DMA, tracked by ASYNCcnt (gfx1250 async-to-LDS path)
__device__ __forceinline__ void async_cp16(unsigned int lds_addr, const float* g) {
  asm volatile("global_load_async_to_lds_b128 %0, %1, off"
               :: "v"(lds_addr), "v"(g)
               : "memory");
}

template <int N>
__device__ __forceinline__ void wait_async() {
#if __has_builtin(__builtin_amdgcn_s_wait_asynccnt)
  __builtin_amdgcn_s_wait_asynccnt((short)N);
#else
  asm volatile("s_wait_asynccnt %0" :: "i"(N) : "memory");
#endif
}

__device__ __forceinline__ void issue_chunk(int j, int b, int tid,
                                            const float* __restrict__ keys,
                                            const float* __restrict__ values,
                                            float* KSl, float* Vsl) {
  const int tI = j >> 5, hI = j & 31;
  const long kbase = (long)(b * T_ + tI) * CK_ * (long)(H_ * W_) + hI * W_;
  const long vbase = (long)(b * T_ + tI) * CV_ * (long)(H_ * W_) + hI * W_;
#pragma unroll
  for (int r = 0; r < 4; ++r) {            // K chunk: 1024 float4
    const int i = tid + r * 256;
    const int ck = i >> 4, w = (i & 15) << 2;
    async_cp16(LDS_OFF(KSl + ck * K_S + w), keys + kbase + (long)ck * (H_ * W_) + w);
  }
#pragma unroll
  for (int r = 0; r < 16; ++r) {           // V chunk: 4096 float4
    const int i = tid + r * 256;
    const int cv = i >> 4, w = (i & 15) << 2;
    async_cp16(LDS_OFF(Vsl + cv * V_S + w), values + vbase + (long)cv * (H_ * W_) + w);
  }
}

__global__ __launch_bounds__(256) void attn_flash_kernel(
    const float* __restrict__ keys,    // [B,T,CK,H,W]
    const float* __restrict__ values,  // [B,T,CV,H,W]
    const float* __restrict__ qkey,    // [B,CK,H*W]
    float* __restrict__ out)           // [B,2*CV,H,W]
{
  __shared__ float Qt[NT_ * QT_S];       // Q^T tile [n'][ck]
  __shared__ float KS[2][CK_ * K_S];     // K chunk, natural [ck][w], double-buffered
  __shared__ float Vs[2][CV_ * V_S];     // V chunk, natural [cv][w], double-buffered
  __shared__ float ST[NT_ * ST_S];       // S^T / P^T  [n'][m]
  __shared__ float rmax[NT_], rsum[NT_], sclb[NT_];

  const int tid  = threadIdx.x;
  const int lane = tid & 31;
  const int wv   = tid >> 5;       // 8 waves
  const int hi   = lane >> 4;      // lane half -> k+2,k+3 slots of frags
  const int l15  = lane & 15;

  const int b  = blockIdx.x / (N_ / NT_);
  const int qt = blockIdx.x % (N_ / NT_);
  const int n0 = qt * NT_;

  // kick off chunk 0 DMA immediately
  issue_chunk(0, b, tid, keys, values, KS[0], Vs[0]);

  // ---- load Q tile transposed: Qt[n'][ck]
  for (int i = tid; i < CK_ * NT_; i += 256) {
    const int ck = i >> 5, nn = i & 31;
    Qt[nn * QT_S + ck] = qkey[(long)(b * CK_ + ck) * N_ + n0 + nn];
  }
  if (tid < NT_) { rmax[tid] = -3.0e38f; rsum[tid] = 0.0f; }

  // tile ownership: S^T tiles (tn = wv&1, tm = wv>>1); PV tiles tn x tc[4*(wv>>1)..+4)
  const int tn   = wv & 1;
  const int tcb  = (wv >> 1) * 4;
  const int tm   = wv >> 1;
  const int ncol = tn * 16 + l15;
  const int nrow = tn * 16 + l15;
  v8f acc[4];
#pragma unroll
  for (int q = 0; q < 4; ++q)
#pragma unroll
    for (int r = 0; r < 8; ++r) acc[q][r] = 0.0f;

  for (int j = 0; j < NCHUNK_; ++j) {
    const int cur = j & 1;
    // prefetch next chunk into the other buffer, then wait for chunk j to land.
    // async loads complete in order, so <= ASYNC_PER_CHUNK outstanding means
    // everything this wave issued for chunk j is in LDS.
    if (j + 1 < NCHUNK_) {
      issue_chunk(j + 1, b, tid, keys, values, KS[1 - cur], Vs[1 - cur]);
      wait_async<ASYNC_PER_CHUNK>();
    } else {
      wait_async<0>();
    }
    __syncthreads();   // barrier A: chunk j visible from every wave

    // ---- S^T = Q^T x K : 32x64 scores, 8 tiles of 16x16, one per wave
    {
      const float* Kb = KS[cur];
      v8f s;
#pragma unroll
      for (int r = 0; r < 8; ++r) s[r] = 0.0f;
#pragma unroll
      for (int kk = 0; kk < 16; ++kk) {
        const int k0 = kk * 4 + 2 * hi;
        v2f a, bb;
        a.x  = Qt[nrow * QT_S + k0];            // A[n][k], A[n][k+1]
        a.y  = Qt[nrow * QT_S + k0 + 1];
        bb.x = Kb[k0 * K_S + tm * 16 + l15];    // B[k][m], B[k+1][m]
        bb.y = Kb[(k0 + 1) * K_S + tm * 16 + l15];
        s = wmma4(a, bb, s);
      }
      // D layout: vgpr r -> row r (+8 for hi lanes), col = lane&15
#pragma unroll
      for (int r = 0; r < 8; ++r)
        ST[(tn * 16 + r + 8 * hi) * ST_S + tm * 16 + l15] = s[r];
    }
    __syncthreads();   // barrier B: S^T complete

    // ---- online softmax over m: wave 0, one query row per lane (contiguous scan)
    if (tid < NT_) {
      const int c = tid;
      float cmax = -3.0e38f;
      for (int m = 0; m < MC_; ++m) cmax = fmaxf(cmax, ST[c * ST_S + m]);
      const float omax = rmax[c];
      const float nmax = fmaxf(omax, cmax);
      const float sc   = __expf(omax - nmax);
      float csum = 0.0f;
      for (int m = 0; m < MC_; ++m) {
        const float p = __expf(ST[c * ST_S + m] - nmax);
        ST[c * ST_S + m] = p;                   // S^T -> P^T in place
        csum += p;
      }
      rmax[c] = nmax;
      rsum[c] = rsum[c] * sc + csum;
      sclb[c] = sc;
    }
    __syncthreads();   // barrier C: P^T + scale ready

    // ---- acc = acc*scale + V x P  (Cv x 32, 4 tiles per wave, K = 64)
    {
      const float* Vb = Vs[cur];
      const float sc = sclb[ncol];
#pragma unroll
      for (int q = 0; q < 4; ++q)
#pragma unroll
        for (int r = 0; r < 8; ++r) acc[q][r] *= sc;
#pragma unroll
      for (int kk = 0; kk < 16; ++kk) {
        const int k0 = kk * 4 + 2 * hi;
        v2f bb;
        bb.x = ST[ncol * ST_S + k0];            // P[k][n], P[k+1][n]
        bb.y = ST[ncol * ST_S + k0 + 1];
#pragma unroll
        for (int q = 0; q < 4; ++q) {
          const int cvrow = (tcb + q) * 16 + l15;
          v2f a;
          a.x = Vb[cvrow * V_S + k0];           // A[cv][m], A[cv][m+1]
          a.y = Vb[cvrow * V_S + k0 + 1];
          acc[q] = wmma4(a, bb, acc[q]);
        }
      }
    }
    __syncthreads();   // barrier D: buffer 1-cur free for next issue
  }

  // ---- epilogue: normalize by running sum, write memory half of output
  const float inv = 1.0f / rsum[ncol];
  const int n = n0 + ncol;
#pragma unroll
  for (int q = 0; q < 4; ++q) {
#pragma unroll
    for (int r = 0; r < 8; ++r) {
      const int cv = (tcb + q) * 16 + r + 8 * hi;
      out[(long)(b * (2 * CV_) + CV_ + cv) * N_ + n] = acc[q][r] * inv;
    }
  }
}

// bilinear(64x128 -> 32x64, align_corners=False) == exact 2x2 average pool
__global__ __launch_bounds__(256) void pool_kernel(const float* __restrict__ x,
                                                   float* __restrict__ out) {
  const int idx = blockIdx.x * 256 + threadIdx.x;
  if (idx >= B_ * CV_ * H_ * W_) return;
  const int p  = idx & (N_ - 1);
  const int c  = (idx >> 11) & (CV_ - 1);
  const int b  = idx >> 19;            // CV_*N_ = 2^19
  const int oh = p >> 6, ow = p & 63;
  const float* base = x + ((long)(b * CV_ + c) * 64 + 2 * oh) * 128 + 2 * ow;
  const float v = 0.25f * (base[0] + base[1] + base[128] + base[129]);
  out[(long)(b * (2 * CV_) + c) * N_ + p] = v;
}

extern "C" void kernel_launch(void* const* d_in, const int* in_sizes, int n_in,
                              void* d_out, int out_size, void* d_ws, size_t ws_size,
                              hipStream_t stream) {
  const float* query_value = (const float*)d_in[0];  // [2,256,64,128]
  const float* mem_keys    = (const float*)d_in[1];  // [2,4,64,32,64]
  const float* mem_vals    = (const float*)d_in[2];  // [2,4,256,32,64]
  const float* query_key   = (const float*)d_in[3];  // [2,64,32,64]
  float* out = (float*)d_out;                        // [2,512,32,64]

  attn_flash_kernel<<<dim3(B_ * (N_ / NT_)), dim3(256), 0, stream>>>(
      mem_keys, mem_vals, query_key, out);
  pool_kernel<<<dim3((B_ * CV_ * H_ * W_ + 255) / 256), dim3(256), 0, stream>>>(
      query_value, out);
}